// ALiBiPositionalEmbeddings_14370960573069
// MI455X (gfx1250) — compile-verified
//
#include <hip/hip_runtime.h>
#include <hip/hip_bf16.h>
#include <stdint.h>

#define LSEQ 1024
#define ROWS 16
#define TPB  256
#define NH   8

typedef unsigned int u32;
typedef __attribute__((ext_vector_type(4))) u32   v4u;
typedef __attribute__((ext_vector_type(8))) int   v8i;
typedef __attribute__((ext_vector_type(4))) int   v4i;
typedef __attribute__((ext_vector_type(4))) float v4f;

// ---------------------------------------------------------------------------
// Kernel 1: per-(b, j) column metadata.
//   c.x = colbase = j + s2[b,j] + s3col[b,j]
//   c.y = addEq   = (occ>102 ? occ : 0) + (response==1 ? 1 : 0)   (gated by eq)
//   c.z = s3col   (subtracted on the diagonal i==j)
//   c.w = bitcast(diff[b,j])
// All terms are integer-valued -> every f32 sum below is exact.
// ---------------------------------------------------------------------------
__global__ __launch_bounds__(TPB) void alibi_prep_kernel(
    const int* __restrict__ diff, const int* __restrict__ resp,
    v4f* __restrict__ col)
{
  const int b = blockIdx.x;
  __shared__ int sdiff[LSEQ];
  for (int j = threadIdx.x; j < LSEQ; j += TPB) sdiff[j] = diff[b * LSEQ + j];
  __syncthreads();
  for (int j = threadIdx.x; j < LSEQ; j += TPB) {
    const int d = sdiff[j];
    int occ = 0;                       // cumsum(eq)*eq collapses to occurrence idx
    for (int k = 0; k <= j; ++k) occ += (sdiff[k] == d) ? 1 : 0;
    const int r = resp[b * LSEQ + j];
    const int p = (r > -1) ? 1 : 0;    // always 1 for these inputs, kept faithful
    const float de2 = (float)(101 - d);
    const float s2  = (de2 > 50.5f) ? de2 : 0.0f;
    const float dox = (r == 1) ? (float)(101 - d * p) : (float)(d * p);
    const float s3  = (dox > 50.5f) ? dox : 0.0f;
    const float addeq = ((occ > 102) ? (float)occ : 0.0f) + ((r == 1) ? 1.0f : 0.0f);
    v4f c;
    c.x = ((float)j + s2) + s3;
    c.y = addeq;
    c.z = s3;
    c.w = __int_as_float(d);
    col[b * LSEQ + j] = c;
  }
}

// ---------------------------------------------------------------------------
// Kernel 2: stream 256 MB of output. TDM pulls the 16 KB column table into
// LDS; each thread owns 4 consecutive columns and writes one NT B128 store
// per (row, head). S/5 is computed once and reused across all 8 heads
// (slopes are exact powers of two -> bit-identical to the reference).
// ---------------------------------------------------------------------------
__global__ __launch_bounds__(TPB) void alibi_mask_kernel(
    const v4f* __restrict__ col, const float* __restrict__ slopes,
    float* __restrict__ out)
{
  const int b  = blockIdx.y;
  const int i0 = blockIdx.x * ROWS;
  __shared__ v4f scol[LSEQ];                       // 16 KB

#if __has_builtin(__builtin_amdgcn_tensor_load_to_lds) && __has_builtin(__builtin_amdgcn_s_wait_tensorcnt)
  if (threadIdx.x < 32) {                          // wave 0 issues the DMA
    const unsigned long long ga =
        (unsigned long long)(uintptr_t)(const void*)(col + (size_t)b * LSEQ);
    // Generic pointers to __shared__ = LDS aperture | byte offset; low 32 bits
    // are the LDS address required by D# group0.
    const u32 lds = (u32)(uintptr_t)(void*)scol;
    v4u g0;
    g0.x = 1u;                                     // count=1, valid user D#
    g0.y = lds;                                    // lds_addr
    g0.z = (u32)ga;                                // global_addr[31:0]
    g0.w = ((u32)(ga >> 32) & 0x01FFFFFFu) | (2u << 30); // ga[56:32] | type=2
    v8i g1;
    g1[0] = (int)(2u << 16);                       // data_size=2 (4B), no multicast
    g1[1] = (int)(4096u << 16);                    // tensor_dim0 = 4096 dwords
    g1[2] = (int)(1u << 16);                       // tensor_dim1 = 1
    g1[3] = (int)(4096u << 16);                    // tile_dim0 = 4096 dwords (16 KB)
    g1[4] = 1;                                     // tile_dim1 = 1, tile_dim2 = 0
    g1[5] = 4096;                                  // tensor_dim0_stride
    g1[6] = 0;
    g1[7] = 0;
    const v4i gz  = {0, 0, 0, 0};                  // groups 2/3 unused (<=2D)
    const v8i gz8 = {0, 0, 0, 0, 0, 0, 0, 0};      // extra group slot (clang-23 6-arg form)
    __builtin_amdgcn_tensor_load_to_lds(g0, g1, gz, gz, gz8, 0);
    __builtin_amdgcn_s_wait_tensorcnt(0);
  }
#else
  for (int j = threadIdx.x; j < LSEQ; j += TPB) scol[j] = col[(size_t)b * LSEQ + j];
#endif
  __syncthreads();

  float sl[NH];
#pragma unroll
  for (int h = 0; h < NH; ++h) sl[h] = slopes[h];

  const int t  = (int)threadIdx.x;
  const int j0 = 4 * t;
  const v4f c0 = scol[j0 + 0];
  const v4f c1 = scol[j0 + 1];
  const v4f c2 = scol[j0 + 2];
  const v4f c3 = scol[j0 + 3];
  const int d0 = __float_as_int(c0.w), d1 = __float_as_int(c1.w);
  const int d2 = __float_as_int(c2.w), d3 = __float_as_int(c3.w);
  const float NEG_INF = -__builtin_huge_valf();

  const int* sdif = (const int*)scol;              // diff[b,i] = sdif[4*i+3]
  float* outb = out + (size_t)b * NH * (size_t)LSEQ * LSEQ + (size_t)j0;
  const size_t plane = (size_t)LSEQ * (size_t)LSEQ;

  for (int i = i0; i < i0 + ROWS; ++i) {
    const int di = sdif[4 * i + 3];
    float S0 = c0.x + ((d0 == di) ? c0.y : 0.0f) - ((j0 + 0 == i) ? c0.z : 0.0f);
    float S1 = c1.x + ((d1 == di) ? c1.y : 0.0f) - ((j0 + 1 == i) ? c1.z : 0.0f);
    float S2 = c2.x + ((d2 == di) ? c2.y : 0.0f) - ((j0 + 2 == i) ? c2.z : 0.0f);
    float S3 = c3.x + ((d3 == di) ? c3.y : 0.0f) - ((j0 + 3 == i) ? c3.z : 0.0f);
    // causal mask folded into S: -inf survives /5 and the positive slope mul
    S0 = (j0 + 0 <= i) ? S0 : NEG_INF;
    S1 = (j0 + 1 <= i) ? S1 : NEG_INF;
    S2 = (j0 + 2 <= i) ? S2 : NEG_INF;
    S3 = (j0 + 3 <= i) ? S3 : NEG_INF;
    // single rounding point (matches reference's final /5.0)
    S0 = S0 / 5.0f; S1 = S1 / 5.0f; S2 = S2 / 5.0f; S3 = S3 / 5.0f;
    float* rowp = outb + (size_t)i * LSEQ;
#pragma unroll
    for (int h = 0; h < NH; ++h) {
      v4f v;
      v.x = S0 * sl[h]; v.y = S1 * sl[h]; v.z = S2 * sl[h]; v.w = S3 * sl[h];
      // 256 MB streaming output > 192 MB L2: bypass with NT stores
      __builtin_nontemporal_store(v, (v4f*)(rowp + (size_t)h * plane));
    }
  }
}

extern "C" void kernel_launch(void* const* d_in, const int* in_sizes, int n_in,
                              void* d_out, int out_size, void* d_ws, size_t ws_size,
                              hipStream_t stream) {
  // setup_inputs order: tensor (unused), slopes, diff, response
  const float* slopes = (const float*)d_in[1];
  const int*   diff   = (const int*)d_in[2];
  const int*   resp   = (const int*)d_in[3];
  const int B = in_sizes[2] / LSEQ;                // = 8

  v4f* col = (v4f*)d_ws;                           // B*1024*16 B = 128 KB scratch

  alibi_prep_kernel<<<dim3(B), TPB, 0, stream>>>(diff, resp, col);
  alibi_mask_kernel<<<dim3(LSEQ / ROWS, B), TPB, 0, stream>>>(
      col, slopes, (float*)d_out);
}